// GCNN_68178310857464
// MI455X (gfx1250) — compile-verified
//
#include <hip/hip_runtime.h>

typedef __attribute__((ext_vector_type(2))) float v2f;
typedef __attribute__((ext_vector_type(4))) float v4f;
typedef __attribute__((ext_vector_type(8))) float v8f;
typedef __attribute__((ext_vector_type(4))) int   v4i;

#define KC 64           // K-chunk staged in LDS per pipeline stage
#define BPAD 4          // LDS row pad: stride 68 words -> conflict-free b64 reads

// --- async global->LDS helpers (gfx1250 ASYNCcnt path, with sync fallback) ---
// Builtin signature (from hipcc diagnostic): param0 is AS1 (global) int4*,
// so it is (global_src, lds_dst, imm offset, imm cpol) with typed pointers.
__device__ __forceinline__ void stage16(const float* gsrc, float* ldst) {
#if __has_builtin(__builtin_amdgcn_global_load_async_to_lds_b128)
    __builtin_amdgcn_global_load_async_to_lds_b128(
        (__attribute__((address_space(1))) v4i*)gsrc,
        (__attribute__((address_space(3))) v4i*)ldst,
        /*offset=*/0, /*cpol=*/0);
#else
    *(v4f*)ldst = *(const v4f*)gsrc;
#endif
}
__device__ __forceinline__ void wait_async() {
#if __has_builtin(__builtin_amdgcn_global_load_async_to_lds_b128)
#if __has_builtin(__builtin_amdgcn_s_wait_asynccnt)
    __builtin_amdgcn_s_wait_asynccnt(0);
#else
    asm volatile("s_wait_asynccnt 0" ::: "memory");
#endif
#endif
}

// ---------------------------------------------------------------------------
// Y[m, o] = sum_k X[m, k] * W[o, k] + bias[o]      (Y = X @ W^T + b)
// f32 WMMA 16x16x4 (exact f32 math). Wave -> 32x32 output tile (2x2 WMMA
// tiles, fragments reused 2x). Block = 256 thr = 8 waves -> 128x64 tile.
// B (weights) double-buffered through LDS via async global->LDS b128 copies;
// A streamed from global (X is L2-resident at 41 MB).
// Requires D % 64 == 0 (D = 512 here).
//
// Fragment layouts (ISA 7.12.2, wave32):
//   A 16x4  f32: lane l holds row m=l&15, ks k0+2*(l>>4)+{0,1}  -> v2f
//   B 4x16  f32: lane l holds col n=l&15, same k pair           -> v2f
//   C/D 16x16:   lane l col n=l&15, reg j <-> row j+8*(l>>4)    -> v8f
// Out-of-range A rows are CLAMPED (row-independent math, never stored), so
// EXEC stays all-ones (WMMA requirement) with no selects in the K loop.
// LDS bank math: word addr = c*68 + k, lanes read banks 4*lo + 2*hi + {0,1}
// -> all 64 banks distinct, conflict-free.
// ---------------------------------------------------------------------------
__global__ __launch_bounds__(256) void gemm_bias_wmma(
    const float* __restrict__ X, const float* __restrict__ W,
    const float* __restrict__ bias, float* __restrict__ Y,
    int rows, int D)
{
    __shared__ float Bsh[2][64][KC + BPAD];

    const int lane = threadIdx.x & 31;
    const int wave = threadIdx.x >> 5;
    const int hi   = lane >> 4;     // 0 or 1
    const int lo   = lane & 15;

    const int ob0 = blockIdx.x * 64;                     // block col base
    const int mb  = blockIdx.y * 128 + (wave >> 1) * 32; // wave row base
    const int cw  = (wave & 1) * 32;                     // wave col base in tile

    const int m0 = mb + lo;
    const int m1 = mb + 16 + lo;
    const int c0 = m0 < rows ? m0 : rows - 1;            // clamp (safe, see top)
    const int c1 = m1 < rows ? m1 : rows - 1;
    const float* xr0 = X + (size_t)c0 * D;
    const float* xr1 = X + (size_t)c1 * D;

    // staging assignment: thread t copies 16B: col = t>>2, seg = (t&3)*4
    const int sc  = threadIdx.x >> 2;
    const int ss  = (threadIdx.x & 3) * 4;
    const float* wrow = W + (size_t)(ob0 + sc) * D + ss;

    const int cb0 = cw + lo;        // B fragment cols (LDS row index)
    const int cb1 = cw + 16 + lo;
    const int koff = hi * 2;

    v8f acc00 = {}, acc01 = {}, acc10 = {}, acc11 = {};

    // prologue: stage chunk 0
    stage16(wrow, &Bsh[0][sc][ss]);
    wait_async();
    __syncthreads();

    int buf = 0;
    for (int kb = 0; kb < D; kb += KC) {
        if (kb + KC < D)                                  // prefetch next chunk
            stage16(wrow + kb + KC, &Bsh[buf ^ 1][sc][ss]);

#pragma unroll
        for (int s = 0; s < KC; s += 4) {
            const int k  = kb + s + koff;                 // global k (A)
            const int kl = s + koff;                      // local k (B in LDS)
            v2f a0 = *(const v2f*)(xr0 + k);
            v2f a1 = *(const v2f*)(xr1 + k);
            v2f b0 = *(const v2f*)&Bsh[buf][cb0][kl];
            v2f b1 = *(const v2f*)&Bsh[buf][cb1][kl];
            acc00 = __builtin_amdgcn_wmma_f32_16x16x4_f32(false, a0, false, b0,
                                                          (short)0, acc00, false, false);
            acc01 = __builtin_amdgcn_wmma_f32_16x16x4_f32(false, a0, false, b1,
                                                          (short)0, acc01, false, false);
            acc10 = __builtin_amdgcn_wmma_f32_16x16x4_f32(false, a1, false, b0,
                                                          (short)0, acc10, false, false);
            acc11 = __builtin_amdgcn_wmma_f32_16x16x4_f32(false, a1, false, b1,
                                                          (short)0, acc11, false, false);
        }
        wait_async();          // own prefetch into buf^1 landed
        __syncthreads();       // everyone done reading buf / staging buf^1
        buf ^= 1;
    }

    const int ob = ob0 + cw;
    const float bv0 = bias[ob + lo];
    const float bv1 = bias[ob + 16 + lo];

#pragma unroll
    for (int j = 0; j < 8; ++j) {
        const int mA = mb + j + 8 * hi;        // row tile 0
        const int mB = mb + 16 + j + 8 * hi;   // row tile 1
        if (mA < rows) {
            Y[(size_t)mA * D + ob + lo]      = acc00[j] + bv0;
            Y[(size_t)mA * D + ob + 16 + lo] = acc01[j] + bv1;
        }
        if (mB < rows) {
            Y[(size_t)mB * D + ob + lo]      = acc10[j] + bv0;
            Y[(size_t)mB * D + ob + 16 + lo] = acc11[j] + bv1;
        }
    }
}

// ---------------------------------------------------------------------------
// For relation-matrix pass q (0..2R-1), H holds rows [r0, r1) of
// H_q = X @ W_q^T + b_q.  Edge (head, r, tail):
//   q <  R (rel==q):   out[head] += H_q[tail]
//   q >= R (rel==q-R): out[tail] += H_q[head]
// One block per edge; non-matching edges exit immediately.
// ---------------------------------------------------------------------------
__global__ void scatter_add(const float* __restrict__ H,
                            const int* __restrict__ heads,
                            const int* __restrict__ tails,
                            const int* __restrict__ rel,
                            float* __restrict__ out,
                            int D, int q, int R, int r0, int r1)
{
    const int e   = blockIdx.x;
    const bool fwd = (q < R);
    const int want = fwd ? q : q - R;
    if (rel[e] != want) return;
    const int src = fwd ? tails[e] : heads[e];
    if (src < r0 || src >= r1) return;
    const int dst = fwd ? heads[e] : tails[e];

    const float* hsrc = H   + (size_t)(src - r0) * D;
    float*       odst = out + (size_t)dst * D;
    for (int d = threadIdx.x; d < D; d += blockDim.x)
        atomicAdd(&odst[d], hsrc[d]);
}

// ---------------------------------------------------------------------------
__global__ void relu4(float* __restrict__ out, size_t nvec)
{
    size_t i = (size_t)blockIdx.x * blockDim.x + threadIdx.x;
    if (i >= nvec) return;
    v4f v = ((const v4f*)out)[i];
#pragma unroll
    for (int j = 0; j < 4; ++j) v[j] = v[j] > 0.0f ? v[j] : 0.0f;
    ((v4f*)out)[i] = v;
}

// ---------------------------------------------------------------------------
extern "C" void kernel_launch(void* const* d_in, const int* in_sizes, int n_in,
                              void* d_out, int out_size, void* d_ws, size_t ws_size,
                              hipStream_t stream)
{
    const float* X     = (const float*)d_in[0];   // [N, D]
    const int*   heads = (const int*)  d_in[1];   // [E]
    const int*   tails = (const int*)  d_in[2];   // [E]
    const int*   rel   = (const int*)  d_in[3];   // [E]
    const float* Wself = (const float*)d_in[4];   // [D, D]
    const float* bself = (const float*)d_in[5];   // [D]
    const float* Wrel  = (const float*)d_in[6];   // [2R, D, D]
    const float* brel  = (const float*)d_in[7];   // [2R, D]
    float* out = (float*)d_out;                   // [N, D]

    const int D  = in_sizes[5];
    const int N  = in_sizes[0] / D;
    const int E  = in_sizes[1];
    const int R2 = in_sizes[7] / D;   // 2R
    const int R  = R2 / 2;

    const dim3 blk(256);

    // 1) out = X @ Wself^T + bself  (writes every element of d_out)
    {
        dim3 g(D / 64, (N + 127) / 128);
        gemm_bias_wmma<<<g, blk, 0, stream>>>(X, Wself, bself, out, N, D);
    }

    // 2) per relation-matrix pass; H_q staged through workspace (L2-resident
    //    at 41 MB for full N), chunked if the workspace is smaller.
    float* Hbuf = (float*)d_ws;
    int chunk = (int)(ws_size / ((size_t)D * sizeof(float)));
    if (chunk > N) chunk = N;
    if (chunk < 1) chunk = 1;

    for (int q = 0; q < R2; ++q) {
        const float* Wq = Wrel + (size_t)q * D * D;
        const float* bq = brel + (size_t)q * D;
        for (int r0 = 0; r0 < N; r0 += chunk) {
            const int nrows = (N - r0) < chunk ? (N - r0) : chunk;
            dim3 g(D / 64, (nrows + 127) / 128);
            gemm_bias_wmma<<<g, blk, 0, stream>>>(X + (size_t)r0 * D, Wq, bq,
                                                  Hbuf, nrows, D);
            scatter_add<<<dim3(E), blk, 0, stream>>>(Hbuf, heads, tails, rel,
                                                     out, D, q, R, r0, r0 + nrows);
        }
    }

    // 3) ReLU in place (N*D divisible by 4 since D % 64 == 0)
    const size_t nvec = (size_t)N * D / 4;
    relu4<<<dim3((unsigned)((nvec + 255) / 256)), blk, 0, stream>>>(out, nvec);
}